// outcomeAttnDecoder_11562051961203
// MI455X (gfx1250) — compile-verified
//
#include <hip/hip_runtime.h>
#include <hip/hip_bf16.h>

#define H 1024
#define V 50257
#define L 512

typedef __attribute__((ext_vector_type(2))) float v2f;
typedef __attribute__((ext_vector_type(8))) float v8f;

// ---------------------------------------------------------------------------
// Generic f32 GEMV on the WMMA pipe:  y = act( W1 @ x1 [+ W2 @ x2] + b1 [+ b2] )
// One wave handles 16 output rows. A = 16x4 weight tile (each row streamed
// linearly: lane m covers K=k0..k0+1, lane m+16 covers K=k0+2..k0+3 of row m).
// B = activation segment broadcast to all 16 columns, so every column of D is
// the same 16 partial sums; C carries the f32 accumulator (bias preloaded).
// Exact f32 math — identical FMA semantics to a scalar GEMV.
// ---------------------------------------------------------------------------
__global__ void __launch_bounds__(128) gemv_wmma(
    const float* __restrict__ W1, const float* __restrict__ x1, int K1,
    const float* __restrict__ W2, const float* __restrict__ x2, int K2,
    const float* __restrict__ b1, const float* __restrict__ b2,
    float* __restrict__ y, int nrows, int act)
{
    const int lane = threadIdx.x & 31;
    const int wave = threadIdx.x >> 5;
    const int row_base = (blockIdx.x * 4 + wave) * 16;
    if (row_base >= nrows) return;            // uniform per wave: EXEC stays full

    const int half = lane >> 4;               // 0: K=k0..k0+1, 1: K=k0+2..k0+3
    const int mrow = lane & 15;
    const int kofs = half * 2;
    int row = row_base + mrow;
    if (row > nrows - 1) row = nrows - 1;     // clamp tail (store is guarded)

    v8f acc;
#pragma unroll
    for (int v = 0; v < 8; ++v) {             // C[m,n] = bias[m] for all n
        int r = row_base + v + half * 8;
        if (r > nrows - 1) r = nrows - 1;
        float bv = b1 ? b1[r] : 0.0f;
        if (b2) bv += b2[r];
        acc[v] = bv;
    }

    {
        const float* wrow = W1 + (long)row * K1;
        for (int k0 = 0; k0 < K1; k0 += 4) {
            v2f a = *(const v2f*)(wrow + k0 + kofs);
            v2f b = *(const v2f*)(x1 + k0 + kofs);
            acc = __builtin_amdgcn_wmma_f32_16x16x4_f32(
                false, a, false, b, (short)0, acc, false, false);
        }
    }
    if (W2) {
        const float* wrow2 = W2 + (long)row * K2;
        for (int k0 = 0; k0 < K2; k0 += 4) {
            v2f a = *(const v2f*)(wrow2 + k0 + kofs);
            v2f b = *(const v2f*)(x2 + k0 + kofs);
            acc = __builtin_amdgcn_wmma_f32_16x16x4_f32(
                false, a, false, b, (short)0, acc, false, false);
        }
    }

    if (mrow == 0) {                          // lanes 0 and 16 emit 8 rows each
        const int rb = row_base + half * 8;
#pragma unroll
        for (int v = 0; v < 8; ++v) {
            int r = rb + v;
            if (r < nrows) {
                float val = acc[v];
                if (act == 1) val = fmaxf(val, 0.0f);
                y[r] = val;
            }
        }
    }
}

// Build xcat0 = [embedded, h0] and the first half of xcat1 = [embedded, ...]
__global__ void prep_concat(const int* __restrict__ input,
                            const float* __restrict__ emb,
                            const float* __restrict__ hidden,
                            float* __restrict__ xcat0, float* __restrict__ xcat1)
{
    int t = blockIdx.x * blockDim.x + threadIdx.x;      // < 2H
    if (t < H) {
        float e = emb[(long)input[0] * H + t];
        xcat0[t] = e;
        xcat1[t] = e;
    } else if (t < 2 * H) {
        xcat0[t] = hidden[t - H];
    }
}

// softmax over L=512; writes ws copy (for the weighted sum) and the output slot
__global__ void softmax_l(const float* __restrict__ logits,
                          float* __restrict__ aw, float* __restrict__ out_aw)
{
    __shared__ float s[L];
    int t = threadIdx.x;
    float v = logits[t];
    s[t] = v; __syncthreads();
    for (int off = L / 2; off; off >>= 1) {
        if (t < off) s[t] = fmaxf(s[t], s[t + off]);
        __syncthreads();
    }
    float m = s[0]; __syncthreads();
    float e = expf(v - m);
    s[t] = e; __syncthreads();
    for (int off = L / 2; off; off >>= 1) {
        if (t < off) s[t] += s[t + off];
        __syncthreads();
    }
    float w = e / s[0];
    aw[t] = w;
    out_aw[t] = w;
}

// attn_applied[h] = sum_l aw[l] * enc[l, h]  (coalesced over h), into xcat1[H+h]
__global__ void attn_apply(const float* __restrict__ aw,
                           const float* __restrict__ enc,
                           float* __restrict__ dst)
{
    int h = blockIdx.x * blockDim.x + threadIdx.x;      // < H
    float s = 0.0f;
    for (int l = 0; l < L; ++l) s = fmaf(aw[l], enc[(long)l * H + h], s);
    dst[h] = s;
}

// LSTM cell pointwise; gate order i,f,g,o; c0 == h0 == hidden
__global__ void lstm_pointwise(const float* __restrict__ gates,
                               const float* __restrict__ hidden,
                               float* __restrict__ hvec, float* __restrict__ out)
{
    int t = blockIdx.x * blockDim.x + threadIdx.x;      // < H
    float ig = gates[t], fg = gates[H + t], gg = gates[2 * H + t], og = gates[3 * H + t];
    float c0 = hidden[t];
    float si = 1.0f / (1.0f + expf(-ig));
    float sf = 1.0f / (1.0f + expf(-fg));
    float so = 1.0f / (1.0f + expf(-og));
    float c = sf * c0 + si * tanhf(gg);
    float h = so * tanhf(c);
    hvec[t] = h;
    out[V + t] = h;            // h output
    out[V + H + t] = c;        // c output
}

// log-softmax over V=50257 (3 cheap passes over 200 KB), writes out[0:V]
__global__ void logsoftmax_v(const float* __restrict__ x, float* __restrict__ out)
{
    __shared__ float s[1024];
    int t = threadIdx.x;
    float m = -INFINITY;
    for (int i = t; i < V; i += 1024) m = fmaxf(m, x[i]);
    s[t] = m; __syncthreads();
    for (int off = 512; off; off >>= 1) {
        if (t < off) s[t] = fmaxf(s[t], s[t + off]);
        __syncthreads();
    }
    float gmax = s[0]; __syncthreads();
    float sum = 0.0f;
    for (int i = t; i < V; i += 1024) sum += expf(x[i] - gmax);
    s[t] = sum; __syncthreads();
    for (int off = 512; off; off >>= 1) {
        if (t < off) s[t] += s[t + off];
        __syncthreads();
    }
    float lse = gmax + logf(s[0]);
    for (int i = t; i < V; i += 1024) out[i] = x[i] - lse;
}

static inline int gemv_blocks(int nrows) {
    int waves = (nrows + 15) / 16;
    return (waves + 3) / 4;                   // 4 waves (128 threads) per block
}

extern "C" void kernel_launch(void* const* d_in, const int* in_sizes, int n_in,
                              void* d_out, int out_size, void* d_ws, size_t ws_size,
                              hipStream_t stream)
{
    const int*   input  = (const int*)  d_in[0];
    const float* hidden = (const float*)d_in[1];
    const float* enc    = (const float*)d_in[2];
    const float* emb    = (const float*)d_in[3];
    const float* attn_W = (const float*)d_in[4];
    const float* attn_b = (const float*)d_in[5];
    const float* comb_W = (const float*)d_in[6];
    const float* comb_b = (const float*)d_in[7];
    const float* W_ih   = (const float*)d_in[8];
    const float* W_hh   = (const float*)d_in[9];
    const float* b_ih   = (const float*)d_in[10];
    const float* b_hh   = (const float*)d_in[11];
    const float* out_W  = (const float*)d_in[12];
    const float* out_b  = (const float*)d_in[13];
    float* out = (float*)d_out;               // [V | h(H) | c(H) | attn_w(L)]

    float* ws      = (float*)d_ws;            // ~247 KB of f32 scratch
    float* xcat0   = ws;                      // 2H   = 2048
    float* xcat1   = ws + 2048;               // 2H   = 2048
    float* logitsL = ws + 4096;               // L    = 512
    float* aw      = ws + 4608;               // L    = 512
    float* xcomb   = ws + 5120;               // H    = 1024
    float* gates   = ws + 6144;               // 4H   = 4096
    float* hvec    = ws + 10240;              // H    = 1024
    float* logitsV = ws + 11264;              // V    = 50257

    // 1) embedded + concat buffers
    prep_concat<<<8, 256, 0, stream>>>(input, emb, hidden, xcat0, xcat1);

    // 2) attn_logits = [emb, h0] @ attn_W.T + attn_b         (512 x 2048)
    gemv_wmma<<<gemv_blocks(L), 128, 0, stream>>>(
        attn_W, xcat0, 2 * H, nullptr, nullptr, 0, attn_b, nullptr, logitsL, L, 0);

    // 3) softmax -> attn_weights (ws + output slot)
    softmax_l<<<1, L, 0, stream>>>(logitsL, aw, out + V + 2 * H);

    // 4) attn_applied = aw @ encoder_outputs -> xcat1[H:2H]
    attn_apply<<<H / 256, 256, 0, stream>>>(aw, enc, xcat1 + H);

    // 5) x = relu([emb, attn_applied] @ comb_W.T + comb_b)   (1024 x 2048)
    gemv_wmma<<<gemv_blocks(H), 128, 0, stream>>>(
        comb_W, xcat1, 2 * H, nullptr, nullptr, 0, comb_b, nullptr, xcomb, H, 1);

    // 6) gates = x @ W_ih.T + b_ih + h0 @ W_hh.T + b_hh      (4096 x 1024, x2)
    gemv_wmma<<<gemv_blocks(4 * H), 128, 0, stream>>>(
        W_ih, xcomb, H, W_hh, hidden, H, b_ih, b_hh, gates, 4 * H, 0);

    // 7) LSTM pointwise -> hvec, out[V:V+H]=h, out[V+H:V+2H]=c
    lstm_pointwise<<<H / 256, 256, 0, stream>>>(gates, hidden, hvec, out);

    // 8) logits = h @ out_W.T + out_b                        (50257 x 1024)
    gemv_wmma<<<gemv_blocks(V), 128, 0, stream>>>(
        out_W, hvec, H, nullptr, nullptr, 0, out_b, nullptr, logitsV, V, 0);

    // 9) log-softmax -> out[0:V]
    logsoftmax_v<<<1, 1024, 0, stream>>>(logitsV, out);
}